// SolventGCN_27650999452232
// MI455X (gfx1250) — compile-verified
//
#include <hip/hip_runtime.h>
#include <hip/hip_bf16.h>

typedef __attribute__((ext_vector_type(2))) float v2f;
typedef __attribute__((ext_vector_type(8))) float v8f;

// ---------------- utility kernels ----------------

__global__ void fill_f32(float* __restrict__ p, float v, long long n) {
  long long i = blockIdx.x * (long long)blockDim.x + threadIdx.x;
  if (i < n) p[i] = v;
}

__global__ void deg_count(const int* __restrict__ dst, float* __restrict__ deg, int E) {
  int e = blockIdx.x * blockDim.x + threadIdx.x;
  if (e < E) atomicAdd(&deg[dst[e]], 1.0f);
}

__global__ void rsqrt_inplace(float* __restrict__ p, int n) {
  int i = blockIdx.x * blockDim.x + threadIdx.x;
  if (i < n) p[i] = rsqrtf(p[i]);  // deg >= 1 always (self loop)
}

// ReLU over float4 groups (n4 = total elements / 4)
__global__ void relu_v4(float* __restrict__ p, long long n4) {
  long long i = blockIdx.x * (long long)blockDim.x + threadIdx.x;
  if (i >= n4) return;
  float4 v = *(float4*)(p + (i << 2));
  v.x = fmaxf(v.x, 0.0f); v.y = fmaxf(v.y, 0.0f);
  v.z = fmaxf(v.z, 0.0f); v.w = fmaxf(v.w, 0.0f);
  *(float4*)(p + (i << 2)) = v;
}

// WT[n*K + k] = W[k*Fout + n]  (tiny, run once per weight before its GEMM)
__global__ void transpose_f32(const float* __restrict__ W, float* __restrict__ WT,
                              int K, int Fout) {
  int i = blockIdx.x * blockDim.x + threadIdx.x;
  if (i >= K * Fout) return;
  int k = i / Fout, n = i % Fout;
  WT[n * K + k] = W[i];
}

// ---------------- WMMA f32 GEMM: H = A[N,K] @ B[K,Fout] (+bias, relu) ----------------
// BT is B transposed: [Fout, K] row-major, so both A and B fragments are
// contiguous 8-byte pairs -> global_load_b64.
// One wave32 per 16x16 output tile. N, Fout multiples of 16; K multiple of 4.
__global__ void wmma_gemm_f32(const float* __restrict__ A, const float* __restrict__ BT,
                              const float* __restrict__ bias, float* __restrict__ H,
                              int N, int K, int Fout, int biasRelu) {
  const int lane = threadIdx.x & 31;
  const int tile = blockIdx.x * (blockDim.x >> 5) + (threadIdx.x >> 5);
  const int tilesN = Fout >> 4;
  const int numTiles = (N >> 4) * tilesN;
  if (tile >= numTiles) return;  // wave-uniform: EXEC stays all-ones for WMMA
  const int m0 = (tile / tilesN) << 4;
  const int n0 = (tile % tilesN) << 4;
  const int half = lane >> 4;   // 0: K={0,1} lanes, 1: K={2,3} lanes
  const int l15 = lane & 15;

  v8f acc = {};
  const float* arow = A + (long long)(m0 + l15) * K + half * 2;
  const float* brow = BT + (long long)(n0 + l15) * K + half * 2;
  for (int k = 0; k < K; k += 4) {
    v2f a, b;
    a.x = arow[k];
    a.y = arow[k + 1];
    b.x = brow[k];
    b.y = brow[k + 1];
    acc = __builtin_amdgcn_wmma_f32_16x16x4_f32(
        /*neg_a=*/false, a, /*neg_b=*/false, b,
        /*c_mod=*/(short)0, acc, /*reuse_a=*/false, /*reuse_b=*/false);
  }

  const int mbase = m0 + half * 8;  // C/D layout: row = (lane>>4)*8 + v
#pragma unroll
  for (int v = 0; v < 8; ++v) {
    float r = acc[v];
    if (biasRelu) {
      r += bias[n0 + l15];
      r = fmaxf(r, 0.0f);
    }
    H[(long long)(mbase + v) * Fout + n0 + l15] = r;
  }
}

// ---------------- GCN aggregation ----------------
// out[i][f] = b[f] + h[i][f]*dinv[i]^2  (bias + self-loop fused); float4 per thread.
// lf4 = log2(F/4); n4 = N*F/4.
__global__ void agg_init_v4(const float* __restrict__ h, const float* __restrict__ dinv,
                            const float* __restrict__ bias, float* __restrict__ out,
                            long long n4, int lf4) {
  long long i = blockIdx.x * (long long)blockDim.x + threadIdx.x;
  if (i >= n4) return;
  int node = (int)(i >> lf4);
  int f4 = ((int)i & ((1 << lf4) - 1)) << 2;
  float di = dinv[node];
  float d2 = di * di;
  float4 hv = *(const float4*)(h + (i << 2));
  float4 bv = *(const float4*)(bias + f4);
  float4 r;
  r.x = fmaf(hv.x, d2, bv.x); r.y = fmaf(hv.y, d2, bv.y);
  r.z = fmaf(hv.z, d2, bv.z); r.w = fmaf(hv.w, d2, bv.w);
  *(float4*)(out + (i << 2)) = r;
}

// out[dst][f..f+3] += h[src][f..f+3] * dinv[src]*dinv[dst]
// lf = log2(F/4): threads-per-edge = F/4; blockDim.x = 256; shift/mask only.
__global__ void edge_scatter_v4(const int* __restrict__ src, const int* __restrict__ dst,
                                const float* __restrict__ h, const float* __restrict__ dinv,
                                float* __restrict__ out, int E, int lf) {
  const int tpe = 1 << lf;                 // threads per edge
  const int epb = blockDim.x >> lf;        // edges per block
  long long e = (long long)blockIdx.x * epb + (threadIdx.x >> lf);
  if (e >= E) return;
  const int f4 = (threadIdx.x & (tpe - 1)) << 2;
  const int F = tpe << 2;
  int sn = src[e], dn = dst[e];
  float nrm = dinv[sn] * dinv[dn];
  float4 hv = *(const float4*)(h + (long long)sn * F + f4);
  float* o = out + (long long)dn * F + f4;
  atomicAdd(o + 0, hv.x * nrm);
  atomicAdd(o + 1, hv.y * nrm);
  atomicAdd(o + 2, hv.z * nrm);
  atomicAdd(o + 3, hv.w * nrm);
}

// ---------------- pooling ----------------
// x >= 0 (post-ReLU): segment max == atomicMax on uint bits, 0-init.
__global__ void pool_accum_v4(const float* __restrict__ x, const int* __restrict__ batch,
                              float* __restrict__ mx, float* __restrict__ sum,
                              float* __restrict__ cnt, long long n4, int lf4) {
  long long i = blockIdx.x * (long long)blockDim.x + threadIdx.x;
  if (i >= n4) return;
  int node = (int)(i >> lf4);
  int f4 = ((int)i & ((1 << lf4) - 1)) << 2;
  int F = 1 << (lf4 + 2);
  int g = batch[node];
  float4 v = *(const float4*)(x + (i << 2));
  float* mg = mx + (long long)g * F + f4;
  float* sg = sum + (long long)g * F + f4;
  atomicMax((unsigned int*)(mg + 0), __float_as_uint(v.x));
  atomicMax((unsigned int*)(mg + 1), __float_as_uint(v.y));
  atomicMax((unsigned int*)(mg + 2), __float_as_uint(v.z));
  atomicMax((unsigned int*)(mg + 3), __float_as_uint(v.w));
  atomicAdd(sg + 0, v.x);
  atomicAdd(sg + 1, v.y);
  atomicAdd(sg + 2, v.z);
  atomicAdd(sg + 3, v.w);
  if (f4 == 0) atomicAdd(&cnt[g], 1.0f);
}

__global__ void embed_build(const float* __restrict__ cMx, const float* __restrict__ cSum,
                            const float* __restrict__ cCnt, const float* __restrict__ sMx,
                            const float* __restrict__ sSum, const float* __restrict__ sCnt,
                            float* __restrict__ embed, int G, int CE, int SE) {
  const int tot = 2 * (CE + SE);
  long long i = blockIdx.x * (long long)blockDim.x + threadIdx.x;
  if (i >= (long long)G * tot) return;
  int g = (int)(i / tot), j = (int)(i % tot);
  float v;
  if (j < CE)               v = cMx[(long long)g * CE + j];
  else if (j < 2 * CE)      v = cSum[(long long)g * CE + (j - CE)] / fmaxf(cCnt[g], 1.0f);
  else if (j < 2 * CE + SE) v = sMx[(long long)g * SE + (j - 2 * CE)];
  else                      v = sSum[(long long)g * SE + (j - 2 * CE - SE)] / fmaxf(sCnt[g], 1.0f);
  embed[i] = v;
}

// out[g] = dot(dense[g,:], Wo) + bo  (one wave32 per graph)
__global__ void final_out(const float* __restrict__ dense, const float* __restrict__ Wo,
                          const float* __restrict__ bo, float* __restrict__ out,
                          int G, int D) {
  int g = blockIdx.x * (blockDim.x >> 5) + (threadIdx.x >> 5);
  int lane = threadIdx.x & 31;
  if (g >= G) return;
  float acc = 0.0f;
  for (int j = lane; j < D; j += 32) acc += dense[(long long)g * D + j] * Wo[j];
  for (int off = 16; off > 0; off >>= 1) acc += __shfl_down(acc, off, 32);
  if (lane == 0) out[g] = acc + bo[0];
}

// ---------------- host orchestration ----------------

extern "C" void kernel_launch(void* const* d_in, const int* in_sizes, int n_in,
                              void* d_out, int out_size, void* d_ws, size_t ws_size,
                              hipStream_t stream) {
  const float* c       = (const float*)d_in[0];
  const int*   c_edge  = (const int*)d_in[1];
  const int*   c_batch = (const int*)d_in[2];
  const float* s       = (const float*)d_in[3];
  const int*   s_edge  = (const int*)d_in[4];
  const int*   s_batch = (const int*)d_in[5];
  const float* Wc[3] = {(const float*)d_in[6], (const float*)d_in[8], (const float*)d_in[10]};
  const float* bc[3] = {(const float*)d_in[7], (const float*)d_in[9], (const float*)d_in[11]};
  const float* Ws[3] = {(const float*)d_in[12], (const float*)d_in[14], (const float*)d_in[16]};
  const float* bs[3] = {(const float*)d_in[13], (const float*)d_in[15], (const float*)d_in[17]};
  const float* Wd = (const float*)d_in[18];
  const float* bd = (const float*)d_in[19];
  const float* Wo = (const float*)d_in[20];
  const float* bo = (const float*)d_in[21];

  const int F  = 64;
  const int CE = in_sizes[7];        // 64
  const int SE = in_sizes[13];       // 32
  const int Nc = in_sizes[2];        // 100000
  const int Ns = in_sizes[5];        // 100000
  const int Ec = in_sizes[1] / 2;    // 1250000
  const int Es = in_sizes[4] / 2;    // 1250000
  const int G  = 2048;
  const int DD = in_sizes[19];       // 128
  const int EMB = 2 * (CE + SE);     // 192

  const int lfC  = __builtin_ctz(CE / 4);   // log2(CE/4) = 4
  const int lfS  = __builtin_ctz(SE / 4);   // log2(SE/4) = 3

  float* w = (float*)d_ws;
  float* hC    = w; w += (size_t)Nc * CE;
  float* actC  = w; w += (size_t)Nc * CE;
  float* hS    = w; w += (size_t)Ns * SE;
  float* actS  = w; w += (size_t)Ns * SE;
  float* dinvC = w; w += Nc;
  float* dinvS = w; w += Ns;
  float* cMx   = w; w += (size_t)G * CE;
  float* cSum  = w; w += (size_t)G * CE;
  float* cCnt  = w; w += G;
  float* sMx   = w; w += (size_t)G * SE;
  float* sSum  = w; w += (size_t)G * SE;
  float* sCnt  = w; w += G;
  float* dense = w; w += (size_t)G * DD;
  float* wT    = w; w += (size_t)EMB * DD;  // transposed-weight scratch (max = Wd)

  float* outp  = (float*)d_out;  // [G]
  float* embed = outp + G;       // [G, EMB] (second tuple output)

  const int T = 256;  // 8 wave32 per block
  auto blk1 = [&](long long n) { return (unsigned)((n + T - 1) / T); };

  // degrees -> dinv (self-loop counted via 1.0 init)
  fill_f32<<<blk1(Nc), T, 0, stream>>>(dinvC, 1.0f, Nc);
  deg_count<<<blk1(Ec), T, 0, stream>>>(c_edge + Ec, dinvC, Ec);
  rsqrt_inplace<<<blk1(Nc), T, 0, stream>>>(dinvC, Nc);
  fill_f32<<<blk1(Ns), T, 0, stream>>>(dinvS, 1.0f, Ns);
  deg_count<<<blk1(Es), T, 0, stream>>>(s_edge + Es, dinvS, Es);
  rsqrt_inplace<<<blk1(Ns), T, 0, stream>>>(dinvS, Ns);

  // ---- c branch: 3 GCN layers ----
  {
    const float* xin = c; int K = F;
    const int epb = T >> lfC;  // edges per block in scatter
    for (int L = 0; L < 3; ++L) {
      transpose_f32<<<blk1(K * CE), T, 0, stream>>>(Wc[L], wT, K, CE);
      int tiles = (Nc / 16) * (CE / 16);
      wmma_gemm_f32<<<(tiles + 7) / 8, T, 0, stream>>>(xin, wT, nullptr, hC, Nc, K, CE, 0);
      long long n4 = (long long)Nc * CE / 4;
      agg_init_v4<<<blk1(n4), T, 0, stream>>>(hC, dinvC, bc[L], actC, n4, lfC);
      edge_scatter_v4<<<(Ec + epb - 1) / epb, T, 0, stream>>>(c_edge, c_edge + Ec, hC, dinvC,
                                                              actC, Ec, lfC);
      relu_v4<<<blk1(n4), T, 0, stream>>>(actC, n4);
      xin = actC; K = CE;
    }
  }
  // ---- s branch: 3 GCN layers ----
  {
    const float* xin = s; int K = F;
    const int epb = T >> lfS;
    for (int L = 0; L < 3; ++L) {
      transpose_f32<<<blk1(K * SE), T, 0, stream>>>(Ws[L], wT, K, SE);
      int tiles = (Ns / 16) * (SE / 16);
      wmma_gemm_f32<<<(tiles + 7) / 8, T, 0, stream>>>(xin, wT, nullptr, hS, Ns, K, SE, 0);
      long long n4 = (long long)Ns * SE / 4;
      agg_init_v4<<<blk1(n4), T, 0, stream>>>(hS, dinvS, bs[L], actS, n4, lfS);
      edge_scatter_v4<<<(Es + epb - 1) / epb, T, 0, stream>>>(s_edge, s_edge + Es, hS, dinvS,
                                                              actS, Es, lfS);
      relu_v4<<<blk1(n4), T, 0, stream>>>(actS, n4);
      xin = actS; K = SE;
    }
  }

  // ---- pooling ----
  fill_f32<<<blk1((long long)G * CE), T, 0, stream>>>(cMx, 0.0f, (long long)G * CE);
  fill_f32<<<blk1((long long)G * CE), T, 0, stream>>>(cSum, 0.0f, (long long)G * CE);
  fill_f32<<<blk1(G), T, 0, stream>>>(cCnt, 0.0f, G);
  fill_f32<<<blk1((long long)G * SE), T, 0, stream>>>(sMx, 0.0f, (long long)G * SE);
  fill_f32<<<blk1((long long)G * SE), T, 0, stream>>>(sSum, 0.0f, (long long)G * SE);
  fill_f32<<<blk1(G), T, 0, stream>>>(sCnt, 0.0f, G);
  pool_accum_v4<<<blk1((long long)Nc * CE / 4), T, 0, stream>>>(actC, c_batch, cMx, cSum, cCnt,
                                                                (long long)Nc * CE / 4, lfC);
  pool_accum_v4<<<blk1((long long)Ns * SE / 4), T, 0, stream>>>(actS, s_batch, sMx, sSum, sCnt,
                                                                (long long)Ns * SE / 4, lfS);

  embed_build<<<blk1((long long)G * EMB), T, 0, stream>>>(cMx, cSum, cCnt, sMx, sSum, sCnt,
                                                          embed, G, CE, SE);

  // ---- dense head: relu(embed @ Wd + bd) via WMMA, then final dot ----
  transpose_f32<<<blk1(EMB * DD), T, 0, stream>>>(Wd, wT, EMB, DD);
  {
    int tiles = (G / 16) * (DD / 16);
    wmma_gemm_f32<<<(tiles + 7) / 8, T, 0, stream>>>(embed, wT, bd, dense, G, EMB, DD, 1);
  }
  final_out<<<(G + 7) / 8, T, 0, stream>>>(dense, Wo, bo, outp, G, DD);
}